// WLconv_55731495632942
// MI455X (gfx1250) — compile-verified
//
#include <hip/hip_runtime.h>

typedef __attribute__((ext_vector_type(2))) float v2f;
typedef __attribute__((ext_vector_type(8))) float v8f;

#define C_IN  128
#define HID   256
#define C_OUT 128
#define MT    32     // node rows per block
#define WL_LD 132    // padded LDS stride (132 mod 64 = 4 -> conflict-free column reads)
#define H_LD  260    // padded LDS stride (260 mod 64 = 4)

// ---------------- workspace zeroing (agg + deg) ----------------
__global__ void zero_ws_kernel(float* __restrict__ ws, long long n) {
    long long i = (long long)blockIdx.x * blockDim.x + threadIdx.x;
    long long stride = (long long)gridDim.x * blockDim.x;
    for (; i < n; i += stride) ws[i] = 0.0f;
}

// ---------------- edge scatter: one wave32 per edge ----------------
// lane l handles channels [4l, 4l+3]; gather x[src] row as coalesced 512B,
// scatter-add into agg[dst] (L2-resident) with global_atomic_add_f32.
__global__ void edge_scatter_kernel(const float* __restrict__ x,
                                    const long long* __restrict__ ei,
                                    const float* __restrict__ ea,
                                    float* __restrict__ agg,
                                    float* __restrict__ deg,
                                    int E) {
    int gid  = blockIdx.x * blockDim.x + threadIdx.x;
    int e    = gid >> 5;
    int lane = threadIdx.x & 31;
    if (e >= E) return;
    long long src = ei[e];
    long long dst = ei[(long long)E + e];
    float w = ea[e];
    float4 v = ((const float4*)(x + src * C_IN))[lane];
    float* a = agg + dst * C_IN + lane * 4;
    atomicAdd(a + 0, v.x * w);
    atomicAdd(a + 1, v.y * w);
    atomicAdd(a + 2, v.z * w);
    atomicAdd(a + 3, v.w * w);
    if (lane == 0) atomicAdd(deg + dst, 1.0f);
}

// ---------------- fused WL-finalize + MLP with fp32 WMMA ----------------
__global__ __launch_bounds__(256) void wl_mlp_kernel(
    const float* __restrict__ x,   const float* __restrict__ agg,
    const float* __restrict__ deg,
    const float* __restrict__ W1,  const float* __restrict__ b1,
    const float* __restrict__ W2,  const float* __restrict__ b2,
    float* __restrict__ out, int Nn) {

    __shared__ float wl[MT * WL_LD];   // [32, 128] wl activations (padded)
    __shared__ float hh[MT * H_LD];    // [32, 256] hidden activations (padded)

    const int tid  = threadIdx.x;
    const int base = blockIdx.x * MT;

    // keep the (L2-resident) weight matrices warm
    __builtin_prefetch(W1, 0, 1);
    __builtin_prefetch(W2, 0, 1);

    // ---- stage wl tile: wl = 0.5*(x + agg/max(deg,1)) ----
    for (int i = tid; i < MT * C_IN; i += 256) {
        int r = i >> 7, c = i & (C_IN - 1);
        int node = base + r;
        float v = 0.0f;
        if (node < Nn) {
            float d = fmaxf(deg[node], 1.0f);
            long long off = (long long)node * C_IN + c;
            v = 0.5f * (x[off] + agg[off] / d);
        }
        wl[r * WL_LD + c] = v;
    }
    __syncthreads();

    const int wave = tid >> 5;
    const int lane = tid & 31;
    const int half = lane >> 4;   // lanes 16-31 hold K=k+2,k+3 of A/B frags
    const int l16  = lane & 15;

    // ---- GEMM1: h = relu(wl @ W1 + b1), [32,128] x [128,256] ----
    {
        const int c0 = wave, c1 = wave + 8;           // two 16-col blocks per wave
        v8f acc00, acc01, acc10, acc11;
        const float bias0 = b1[c0 * 16 + l16];
        const float bias1 = b1[c1 * 16 + l16];
        for (int j = 0; j < 8; ++j) {
            acc00[j] = bias0; acc10[j] = bias0;
            acc01[j] = bias1; acc11[j] = bias1;
        }
        for (int k = 0; k < C_IN; k += 4) {
            const int kk = k + 2 * half;
            v2f a0, a1, bb0, bb1;
            a0.x = wl[l16 * WL_LD + kk];          a0.y = wl[l16 * WL_LD + kk + 1];
            a1.x = wl[(16 + l16) * WL_LD + kk];   a1.y = wl[(16 + l16) * WL_LD + kk + 1];
            bb0.x = W1[kk * HID + c0 * 16 + l16]; bb0.y = W1[(kk + 1) * HID + c0 * 16 + l16];
            bb1.x = W1[kk * HID + c1 * 16 + l16]; bb1.y = W1[(kk + 1) * HID + c1 * 16 + l16];
            acc00 = __builtin_amdgcn_wmma_f32_16x16x4_f32(false, a0, false, bb0, (short)0, acc00, false, false);
            acc01 = __builtin_amdgcn_wmma_f32_16x16x4_f32(false, a0, false, bb1, (short)0, acc01, false, false);
            acc10 = __builtin_amdgcn_wmma_f32_16x16x4_f32(false, a1, false, bb0, (short)0, acc10, false, false);
            acc11 = __builtin_amdgcn_wmma_f32_16x16x4_f32(false, a1, false, bb1, (short)0, acc11, false, false);
        }
        // D layout: VGPR j -> row (j + 8*half), col = l16 within the 16x16 tile
        for (int j = 0; j < 8; ++j) {
            const int r0 = j + 8 * half;
            const int r1 = 16 + j + 8 * half;
            hh[r0 * H_LD + c0 * 16 + l16] = fmaxf(acc00[j], 0.0f);
            hh[r0 * H_LD + c1 * 16 + l16] = fmaxf(acc01[j], 0.0f);
            hh[r1 * H_LD + c0 * 16 + l16] = fmaxf(acc10[j], 0.0f);
            hh[r1 * H_LD + c1 * 16 + l16] = fmaxf(acc11[j], 0.0f);
        }
    }
    __syncthreads();

    // ---- GEMM2: out = h @ W2 + b2, [32,256] x [256,128] ----
    {
        const int c = wave;                           // one 16-col block per wave
        v8f acc0, acc1;
        const float bias = b2[c * 16 + l16];
        for (int j = 0; j < 8; ++j) { acc0[j] = bias; acc1[j] = bias; }
        for (int k = 0; k < HID; k += 4) {
            const int kk = k + 2 * half;
            v2f a0, a1, bb;
            a0.x = hh[l16 * H_LD + kk];          a0.y = hh[l16 * H_LD + kk + 1];
            a1.x = hh[(16 + l16) * H_LD + kk];   a1.y = hh[(16 + l16) * H_LD + kk + 1];
            bb.x = W2[kk * C_OUT + c * 16 + l16];
            bb.y = W2[(kk + 1) * C_OUT + c * 16 + l16];
            acc0 = __builtin_amdgcn_wmma_f32_16x16x4_f32(false, a0, false, bb, (short)0, acc0, false, false);
            acc1 = __builtin_amdgcn_wmma_f32_16x16x4_f32(false, a1, false, bb, (short)0, acc1, false, false);
        }
        for (int j = 0; j < 8; ++j) {
            const int n0 = base + j + 8 * half;
            const int n1 = base + 16 + j + 8 * half;
            if (n0 < Nn) out[(long long)n0 * C_OUT + c * 16 + l16] = acc0[j];
            if (n1 < Nn) out[(long long)n1 * C_OUT + c * 16 + l16] = acc1[j];
        }
    }
}

extern "C" void kernel_launch(void* const* d_in, const int* in_sizes, int n_in,
                              void* d_out, int out_size, void* d_ws, size_t ws_size,
                              hipStream_t stream) {
    const float*     x   = (const float*)d_in[0];
    const long long* ei  = (const long long*)d_in[1];   // int64 edge_index [2,E]
    const float*     ea  = (const float*)d_in[2];
    const float*     W1  = (const float*)d_in[3];
    const float*     b1  = (const float*)d_in[4];
    const float*     W2  = (const float*)d_in[5];
    const float*     b2  = (const float*)d_in[6];
    float*           out = (float*)d_out;

    const int Nn = in_sizes[0] / C_IN;   // 50000
    const int E  = in_sizes[2];          // 800000

    float* agg = (float*)d_ws;                       // [N,128]
    float* deg = agg + (size_t)Nn * C_IN;            // [N]

    // zero agg + deg (deterministic each call)
    const long long zn = (long long)Nn * C_IN + Nn;
    zero_ws_kernel<<<2048, 256, 0, stream>>>((float*)d_ws, zn);

    // one wave per edge: E waves -> E*32 threads
    const int sblocks = (E * 32 + 255) / 256;
    edge_scatter_kernel<<<sblocks, 256, 0, stream>>>(x, ei, ea, agg, deg, E);

    // fused finalize + MLP
    const int mblocks = (Nn + MT - 1) / MT;
    wl_mlp_kernel<<<mblocks, 256, 0, stream>>>(x, agg, deg, W1, b1, W2, b2, out, Nn);
}